// CustomLoss_39565238731168
// MI455X (gfx1250) — compile-verified
//
#include <hip/hip_runtime.h>
#include <math.h>

#define NSEED 24
#define BATCH 8192
#define WPB 2                 // waves (=batches) per block
#define TWO_PI 6.283185307179586f

typedef float v2f __attribute__((ext_vector_type(2)));
typedef float v8f __attribute__((ext_vector_type(8)));

// ---- f32 WMMA fragment loaders (CDNA5 ISA 7.12.2, 32-bit layouts) ----
// A 16x4 (MxK): lanes 0-15 hold K=k0+0,k0+1 ; lanes 16-31 hold K=k0+2,k0+3
__device__ inline v2f ldA(const float S[32][33], int rt, int k0, int lm, int half) {
  const int m = rt * 16 + lm;
  const int k = k0 + 2 * half;
  v2f a; a.x = S[m][k]; a.y = S[m][k + 1]; return a;
}
// B 4x16 (KxN): mirrored — lane half selects K pair, lane%16 = column
__device__ inline v2f ldB(const float S[32][33], int k0, int ct, int lm, int half) {
  const int n = ct * 16 + lm;
  const int k = k0 + 2 * half;
  v2f b; b.x = S[k][n]; b.y = S[k + 1][n]; return b;
}
// C/D 16x16 f32: VGPR v -> row v (lanes 0-15) / row v+8 (lanes 16-31), N = lane%16
__device__ inline void stC(float S[32][33], int rt, int ct, int lm, int half, v8f c) {
#pragma unroll
  for (int v = 0; v < 8; ++v)
    S[rt * 16 + half * 8 + v][ct * 16 + lm] = c[v];
}

// acc += A(32xK tile row rt) * B(Kx32 tile col ct), K stepped by 4 via v_wmma_f32_16x16x4_f32
template <int KMAX>
__device__ inline v8f mmAcc(const float A[32][33], const float Bm[32][33],
                            int rt, int ct, int lm, int half) {
  v8f acc = {};
#pragma unroll
  for (int k0 = 0; k0 < KMAX; k0 += 4) {
    v2f a = ldA(A, rt, k0, lm, half);
    v2f b = ldB(Bm, k0, ct, lm, half);
    acc = __builtin_amdgcn_wmma_f32_16x16x4_f32(false, a, false, b, (short)0, acc,
                                                false, false);
  }
  return acc;
}

__global__ __launch_bounds__(WPB * 32) void loss_main(
    const float* __restrict__ outp, const float* __restrict__ tgtp,
    float* __restrict__ ws) {
  __shared__ float sF[2][32][33];        // DFT matrix: re, im (block-shared)
  __shared__ float sX[WPB][2][32][33];   // density re/im; re later reused for pred_dp
  __shared__ float sY[WPB][2][32][33];   // Y = F*X ; later Gaussian A/B factors
  __shared__ float sS[WPB][NSEED * 3];   // this wave's seeds (output, then targets)

  const int tid = threadIdx.x;
  const int w = tid >> 5;
  const int lane = tid & 31;
  const int lm = lane & 15;
  const int half = lane >> 4;
  const int b = blockIdx.x * WPB + w;

  // F[m][k] = exp(-2*pi*i*m*k/32) ; exponent reduced mod 32
  for (int e = tid; e < 32 * 32; e += WPB * 32) {
    int m = e >> 5, k = e & 31;
    float ang = -TWO_PI * (float)((m * k) & 31) * (1.0f / 32.0f);
    float si, co;
    __sincosf(ang, &si, &co);
    sF[0][m][k] = co;
    sF[1][m][k] = si;
  }
  for (int i = lane; i < NSEED * 3; i += 32) sS[w][i] = outp[b * (NSEED * 3) + i];
  __syncthreads();

  // ---- density field: exp(i*2pi*phase) inside radius-8 circle, 1 outside ----
  for (int r = 0; r < 32; ++r) {
    const int c = lane;
    const int dr = r - 16, dc = c - 16;
    float xr = 1.0f, xi = 0.0f;
    if (dr * dr + dc * dc <= 64) {       // inside circle (pad-region label = 0)
      const int i = r - 8, j = c - 8;    // i,j >= 0 guaranteed inside circle
      int lab = 0;
      if (i < 16 && j < 16) {            // argmin Voronoi label over 24 seeds
        float best = 3.4e38f;
#pragma unroll
        for (int s = 0; s < NSEED; ++s) {
          float dx = (float)i - sS[w][3 * s];
          float dy = (float)j - sS[w][3 * s + 1];
          float d = dx * dx + dy * dy;
          if (d < best) { best = d; lab = s; }   // '<' keeps first minimum
        }
      }
      __sincosf(TWO_PI * sS[w][3 * lab + 2], &xi, &xr);
    }
    sX[w][0][r][c] = xr;
    sX[w][1][r][c] = xi;
  }
  __syncthreads();

  // ---- Y = F * X (complex, 4 real 32^3 GEMMs via WMMA) ----
#pragma unroll
  for (int rt = 0; rt < 2; ++rt)
#pragma unroll
    for (int ct = 0; ct < 2; ++ct) {
      v8f rr = mmAcc<32>(sF[0], sX[w][0], rt, ct, lm, half);
      v8f ii = mmAcc<32>(sF[1], sX[w][1], rt, ct, lm, half);
      v8f ri = mmAcc<32>(sF[0], sX[w][1], rt, ct, lm, half);
      v8f ir = mmAcc<32>(sF[1], sX[w][0], rt, ct, lm, half);
      stC(sY[w][0], rt, ct, lm, half, rr - ii);
      stC(sY[w][1], rt, ct, lm, half, ri + ir);
    }
  __syncthreads();

  // ---- M = Y * F ; pred_dp[(m+16)%32][(n+16)%32] = |M[m][n]|  (fftshift folded) ----
#pragma unroll
  for (int rt = 0; rt < 2; ++rt)
#pragma unroll
    for (int ct = 0; ct < 2; ++ct) {
      v8f rr = mmAcc<32>(sY[w][0], sF[0], rt, ct, lm, half);
      v8f ii = mmAcc<32>(sY[w][1], sF[1], rt, ct, lm, half);
      v8f ri = mmAcc<32>(sY[w][0], sF[1], rt, ct, lm, half);
      v8f ir = mmAcc<32>(sY[w][1], sF[0], rt, ct, lm, half);
      v8f mr = rr - ii;
      v8f mi = ri + ir;
#pragma unroll
      for (int v = 0; v < 8; ++v) {
        int m = rt * 16 + half * 8 + v;
        int n = ct * 16 + lm;
        sX[w][0][(m + 16) & 31][(n + 16) & 31] =
            sqrtf(mr[v] * mr[v] + mi[v] * mi[v]);
      }
    }
  __syncthreads();

  // ---- target: separable 24-Gaussian splat as rank-24 GEMM ----
  for (int i = lane; i < NSEED * 3; i += 32) sS[w][i] = tgtp[b * (NSEED * 3) + i];
  __syncthreads();
  const float inv = 1.0f / (2.0f * 1.5f * 1.5f);
  for (int idx = lane; idx < 32 * NSEED; idx += 32) {
    int r = idx / NSEED, s = idx - r * NSEED;        // A[r][s] = amp*exp(-(r-ty)^2/4.5)
    float dy = (float)r - sS[w][3 * s + 1];
    sY[w][0][r][s] = sS[w][3 * s + 2] * __expf(-dy * dy * inv);
    int s2 = idx >> 5, c2 = idx & 31;                // B[s][c] = exp(-(c-tx)^2/4.5)
    float dx = (float)c2 - sS[w][3 * s2];
    sY[w][1][s2][c2] = __expf(-dx * dx * inv);
  }
  __syncthreads();

  // ---- T = A*B (K=24) ; fuse per-batch partial reductions ----
  float sp = 0.f, st = 0.f, spt = 0.f, mxp = 0.f, mxt = 0.f;
#pragma unroll
  for (int rt = 0; rt < 2; ++rt)
#pragma unroll
    for (int ct = 0; ct < 2; ++ct) {
      v8f t = mmAcc<NSEED>(sY[w][0], sY[w][1], rt, ct, lm, half);
#pragma unroll
      for (int v = 0; v < 8; ++v) {
        int m = rt * 16 + half * 8 + v;
        int n = ct * 16 + lm;
        float tt = t[v];
        float pp = sX[w][0][m][n];
        sp += pp * pp;
        st += tt * tt;
        spt += pp * tt;
        mxp = fmaxf(mxp, pp);
        mxt = fmaxf(mxt, tt);
      }
    }
  // wave32 reduction
  for (int off = 16; off > 0; off >>= 1) {
    sp += __shfl_down(sp, off, 32);
    st += __shfl_down(st, off, 32);
    spt += __shfl_down(spt, off, 32);
    mxp = fmaxf(mxp, __shfl_down(mxp, off, 32));
    mxt = fmaxf(mxt, __shfl_down(mxt, off, 32));
  }
  if (lane == 0) {
    float* o = ws + (size_t)b * 5;
    o[0] = sp; o[1] = st; o[2] = spt; o[3] = mxp; o[4] = mxt;
  }
}

// Deterministic final reduce of 8192 x {SP,ST,SPT,maxP,maxT} -> scalar loss.
// loss = (SP*MT/MP + ST*MP/MT - 2*SPT) / sqrt(SP*ST)   (normalization folded)
__global__ __launch_bounds__(256) void loss_final(const float* __restrict__ ws,
                                                  float* __restrict__ out) {
  __shared__ float red[5][256];
  const int tid = threadIdx.x;
  float sp = 0.f, st = 0.f, spt = 0.f, mxp = 0.f, mxt = 0.f;
  for (int b = tid; b < BATCH; b += 256) {
    const float* o = ws + (size_t)b * 5;
    sp += o[0]; st += o[1]; spt += o[2];
    mxp = fmaxf(mxp, o[3]); mxt = fmaxf(mxt, o[4]);
  }
  red[0][tid] = sp; red[1][tid] = st; red[2][tid] = spt;
  red[3][tid] = mxp; red[4][tid] = mxt;
  __syncthreads();
  for (int s = 128; s > 0; s >>= 1) {
    if (tid < s) {
      red[0][tid] += red[0][tid + s];
      red[1][tid] += red[1][tid + s];
      red[2][tid] += red[2][tid + s];
      red[3][tid] = fmaxf(red[3][tid], red[3][tid + s]);
      red[4][tid] = fmaxf(red[4][tid], red[4][tid + s]);
    }
    __syncthreads();
  }
  if (tid == 0) {
    float SP = red[0][0], ST = red[1][0], SPT = red[2][0];
    float MP = red[3][0], MT = red[4][0];
    out[0] = (SP * (MT / MP) + ST * (MP / MT) - 2.0f * SPT) / sqrtf(SP * ST);
  }
}

extern "C" void kernel_launch(void* const* d_in, const int* in_sizes, int n_in,
                              void* d_out, int out_size, void* d_ws, size_t ws_size,
                              hipStream_t stream) {
  (void)in_sizes; (void)n_in; (void)out_size; (void)ws_size;
  const float* outp = (const float*)d_in[0];
  const float* tgtp = (const float*)d_in[1];
  float* ws = (float*)d_ws;                       // needs 8192*5*4 = 160 KB
  loss_main<<<BATCH / WPB, WPB * 32, 0, stream>>>(outp, tgtp, ws);
  loss_final<<<1, 256, 0, stream>>>(ws, (float*)d_out);
}